// SphericalHarmonics_91122026152643
// MI455X (gfx1250) — compile-verified
//
#include <hip/hip_runtime.h>
#include <stdint.h>

#define BLOCK 128
#define LMAX  10
#define NOUT  101   // 100 SH coefficients + time index

namespace {

constexpr double PI_D = 3.14159265358979323846264338327950288;

constexpr double factd(int n) {
    double r = 1.0;
    for (int i = 2; i <= n; ++i) r *= (double)i;
    return r;
}

// constexpr Newton sqrt (handles the tiny K values at l=m=9, ~2.4e-16)
constexpr double csqrt(double x) {
    double g = x > 1.0 ? x : 1.0;
    for (int i = 0; i < 200; ++i) g = 0.5 * (g + x / g);
    return g;
}

struct KArr { float v[LMAX][LMAX]; };

constexpr KArr makeK() {
    KArr a{};
    for (int l = 0; l < LMAX; ++l)
        for (int m = 0; m <= l; ++m) {
            double k = csqrt((2.0 * l + 1.0) / (4.0 * PI_D) * factd(l - m) / factd(l + m));
            if (m > 0) k *= csqrt(2.0);   // fold sqrt(2) for m != 0 into K
            a.v[l][m] = (float)k;
        }
    return a;
}

constexpr KArr KT = makeK();   // folds to immediates after full unroll

} // namespace

__global__ __launch_bounds__(BLOCK)
void sh_embed_kernel(const float* __restrict__ lonlat, float* __restrict__ out, int n) {
    __shared__ float s_in[BLOCK * 3];       // 1.5 KB input stage
    __shared__ float s_out[BLOCK * NOUT];   // 50.5 KB output stage (stride 101: odd -> conflict-free)

    const int tid    = threadIdx.x;
    const int block0 = blockIdx.x * BLOCK;
    const bool full  = (block0 + BLOCK) <= n;

    // ---- Stage input tile into LDS ----
    if (full) {
        // CDNA5 async copy: 96 lanes x 16B = 1536B = BLOCK*3 floats, contiguous
        if (tid < (BLOCK * 3) / 4) {
            uint64_t gaddr = (uint64_t)(uintptr_t)(lonlat + (size_t)block0 * 3) + (uint64_t)tid * 16u;
            uint32_t laddr = (uint32_t)(uintptr_t)(&s_in[tid * 4]);
            asm volatile("global_load_async_to_lds_b128 %0, %1, off"
                         :: "v"(laddr), "v"(gaddr) : "memory");
        }
        asm volatile("s_wait_asynccnt 0x0" ::: "memory");
    } else {
        int valid = n - block0; if (valid < 0) valid = 0;
        for (int k = tid; k < valid * 3; k += BLOCK)
            s_in[k] = lonlat[(size_t)block0 * 3 + k];
    }
    __syncthreads();

    // ---- Per-point spherical harmonics ----
    const int gid = block0 + tid;
    if (gid < n) {
        const float lon = s_in[tid * 3 + 0];
        const float lat = s_in[tid * 3 + 1];
        const float tix = s_in[tid * 3 + 2];

        constexpr float D2R = 0.017453292519943295f;
        const float phi = (lon + 180.0f) * D2R;   // [0, 2pi]
        const float th  = (lat +  90.0f) * D2R;   // [0, pi]

        const float x    = __cosf(th);
        const float s    = __sinf(th);            // >= 0 on [0, pi]
        const float cphi = __cosf(phi);
        const float sphi = __sinf(phi);

        float* row = &s_out[tid * NOUT];

        float cm = 1.0f, sm = 0.0f;   // cos(m*phi), sin(m*phi)
        float pmm = 1.0f;             // P_m^m with Condon-Shortley phase

        auto emit = [&](int l, int m, float p) {
            const float kq = KT.v[l][m] * p;      // compile-time constant K
            const int   c  = l * l + l;
            if (m == 0) {
                row[c] = kq;
            } else {
                row[c + m] = kq * cm;             // m > 0: cos branch
                row[c - m] = kq * sm;             // m < 0: sin branch
            }
        };

#pragma unroll
        for (int m = 0; m < LMAX; ++m) {
            if (m > 0) {
                pmm *= -(2.0f * (float)m - 1.0f) * s;
                float cn = cm * cphi - sm * sphi;
                float sn = sm * cphi + cm * sphi;
                cm = cn; sm = sn;
            }
            emit(m, m, pmm);
            float pm2 = pmm, pm1 = 0.0f;
            if (m + 1 < LMAX) {
                pm1 = x * (2.0f * (float)m + 1.0f) * pmm;
                emit(m + 1, m, pm1);
            }
#pragma unroll
            for (int l = m + 2; l < LMAX; ++l) {
                float p = ((2.0f * (float)l - 1.0f) * x * pm1 - (float)(l + m - 1) * pm2)
                          * (1.0f / (float)(l - m));
                emit(l, m, p);
                pm2 = pm1; pm1 = p;
            }
        }
        row[100] = tix;
    }
    __syncthreads();

    // ---- Coalesced writeout: LDS tile -> global as float4 ----
    if (full) {
        float4*       dst = (float4*)(out + (size_t)block0 * NOUT);  // 51712B tile, 16B aligned
        const float4* src = (const float4*)s_out;
#pragma unroll 4
        for (int k = tid; k < (BLOCK * NOUT) / 4; k += BLOCK)
            dst[k] = src[k];
    } else {
        int valid = n - block0; if (valid < 0) valid = 0;
        const size_t base = (size_t)block0 * NOUT;
        for (int k = tid; k < valid * NOUT; k += BLOCK)
            out[base + k] = s_out[k];
    }
}

extern "C" void kernel_launch(void* const* d_in, const int* in_sizes, int n_in,
                              void* d_out, int out_size, void* d_ws, size_t ws_size,
                              hipStream_t stream) {
    const float* lonlat = (const float*)d_in[0];
    float*       out    = (float*)d_out;
    const int n    = in_sizes[0] / 3;          // (N,3) flattened
    const int grid = (n + BLOCK - 1) / BLOCK;  // 15625 blocks for N=2M
    hipLaunchKernelGGL(sh_embed_kernel, dim3(grid), dim3(BLOCK), 0, stream,
                       lonlat, out, n);
}